// Striatum_25434796327642
// MI455X (gfx1250) — compile-verified
//
#include <hip/hip_runtime.h>

// MI455X / gfx1250, wave32. Masked-GEMM: out[2048,49152] = x[2048,15] @ (w*mask^T) + b
// Output-store bound (403 MB writes @ 23.3 TB/s ~= 17us floor); all inputs L2-resident.
// Math done on the f32 WMMA pipe: V_WMMA_F32_16X16X4_F32, 4 chained ops per 16x16 tile.

typedef __attribute__((ext_vector_type(2))) float v2f;
typedef __attribute__((ext_vector_type(8))) float v8f;

#define BATCH 2048
#define IN_D  15
#define UNITS 49152
#define KPAD  16

// ---- prep: masked, K-padded weights Wp[16][UNITS] (row-major in N) -------------
__global__ void __launch_bounds__(256)
prep_w_kernel(const float* __restrict__ w,     // [15, UNITS]
              const float* __restrict__ mask,  // [UNITS, 15]
              float* __restrict__ Wp) {        // [16, UNITS]
  int n = blockIdx.x * blockDim.x + threadIdx.x;
  if (n >= UNITS) return;
#pragma unroll
  for (int k = 0; k < KPAD; ++k) {
    float v = 0.0f;
    if (k < IN_D) v = w[k * UNITS + n] * mask[n * IN_D + k];
    Wp[(size_t)k * UNITS + n] = v;
  }
}

// ---- prep: K-padded activations Ap[BATCH][16] ----------------------------------
__global__ void __launch_bounds__(256)
prep_a_kernel(const float* __restrict__ x,     // [BATCH, 15]
              float* __restrict__ Ap) {        // [BATCH, 16]
  int i = blockIdx.x * blockDim.x + threadIdx.x;  // over BATCH*16
  int r = i >> 4;
  int k = i & 15;
  Ap[i] = (k < IN_D) ? x[r * IN_D + k] : 0.0f;
}

// ---- main: one wave32 per 16x16 output tile, 4x V_WMMA_F32_16X16X4_F32 ---------
__global__ void __launch_bounds__(256)
striatum_wmma_kernel(const float* __restrict__ Ap,    // [BATCH, 16]
                     const float* __restrict__ Wp,    // [16, UNITS]
                     const float* __restrict__ bias,  // [UNITS]
                     float* __restrict__ out) {       // [BATCH, UNITS]
  const int lane  = threadIdx.x & 31;
  const int wave  = threadIdx.x >> 5;
  const int nTile = blockIdx.x * 8 + wave;   // 8 waves/block along N
  const int mTile = blockIdx.y;

  const int lane16 = lane & 15;
  const int half   = lane >> 4;              // 0: lanes 0-15, 1: lanes 16-31

  // A operand: lane holds row M = mTile*16 + lane16, K-pair = half*2 (+4s)
  const int rowA = mTile * 16 + lane16;
  // B operand: lane holds column N = nTile*16 + lane16, same K-pair split
  const int n    = nTile * 16 + lane16;
  const int kOff = half * 2;

  v8f c = {};
#pragma unroll
  for (int s = 0; s < 4; ++s) {
    const int k = 4 * s + kOff;
    // contiguous 8B-aligned pair -> global_load_b64
    v2f a = *(const v2f*)(Ap + rowA * KPAD + k);
    v2f b;
    b.x = Wp[(size_t)k * UNITS + n];
    b.y = Wp[(size_t)(k + 1) * UNITS + n];
    // (neg_a, A, neg_b, B, c_mod, C, reuse_a, reuse_b)
    c = __builtin_amdgcn_wmma_f32_16x16x4_f32(false, a, false, b,
                                              (short)0, c, false, false);
  }

  const float bv = bias[n];
  // C/D layout: VGPR v -> M = v + 8*half, N = lane16
  const size_t base = (size_t)(mTile * 16 + half * 8) * UNITS + (size_t)n;
#pragma unroll
  for (int v = 0; v < 8; ++v) {
    // write-once 403 MB stream: non-temporal so it doesn't evict Wp/Ap from L2
    __builtin_nontemporal_store(c[v] + bv, out + base + (size_t)v * UNITS);
  }
}

extern "C" void kernel_launch(void* const* d_in, const int* in_sizes, int n_in,
                              void* d_out, int out_size, void* d_ws, size_t ws_size,
                              hipStream_t stream) {
  const float* x    = (const float*)d_in[0];  // [2048, 15]
  const float* w    = (const float*)d_in[1];  // [15, 49152]
  const float* bias = (const float*)d_in[2];  // [49152]
  const float* mask = (const float*)d_in[3];  // [49152, 15]
  float* out = (float*)d_out;

  float* Wp = (float*)d_ws;                                   // 16*49152*4 = 3 MiB
  float* Ap = (float*)((char*)d_ws + (size_t)KPAD * UNITS * 4); // 2048*16*4 = 128 KiB

  prep_w_kernel<<<UNITS / 256, 256, 0, stream>>>(w, mask, Wp);
  prep_a_kernel<<<(BATCH * KPAD) / 256, 256, 0, stream>>>(x, Ap);

  dim3 grid(UNITS / (16 * 8), BATCH / 16);  // (384, 128) blocks, 8 waves each
  striatum_wmma_kernel<<<grid, 256, 0, stream>>>(Ap, Wp, bias, out);
}